// SSIM3D_80358838108257
// MI455X (gfx1250) — compile-verified
//
#include <hip/hip_runtime.h>

// ---------------------------------------------------------------------------
// SSIM3D for MI455X (gfx1250, wave32).
// Pass 1: W-axis blur of all 5 product fields via V_WMMA_F32_16X16X4_F32
//         (banded-GEMM formulation of the 11-tap conv, fp32-exact).
// Pass 2: H-axis blur, register sliding window (coalesced along W).
// Pass 3: D-axis blur + SSIM map + double-precision mean reduction.
// Workspace: bufA(5 fields) + bufB(5 fields) + 1 double accumulator
//            = 10*VOL*4 + 8 = 327,680,008 bytes.
// ---------------------------------------------------------------------------

typedef float v2f __attribute__((ext_vector_type(2)));
typedef float v8f __attribute__((ext_vector_type(8)));

#define NB    2
#define S     160
#define ROWL  S
#define PLANE (S * S)
#define CUBE  (S * S * S)          // 4,096,000
#define VOL   (NB * CUBE)          // 8,192,000

// Gaussian(sigma=1.5), 11 taps, normalized (matches _gaussian_1d exactly in fp32)
__constant__ float GK[11] = {
    0.00102838f, 0.00759866f, 0.03600184f, 0.10936053f, 0.21300555f,
    0.26601174f, 0.21300555f, 0.10936053f, 0.03600184f, 0.00759866f,
    0.00102838f};

// ---------------------------------------------------------------------------
// Pass 1: one wave (32 threads) per (n, d, h-tile of 16). Loads a 16x160 slab
// of img1/img2 (+zero halo) into LDS, then for each 16-wide output tile runs
// 7 K-chunks x 5 fields of V_WMMA_F32_16X16X4_F32.
// ---------------------------------------------------------------------------
#define SLAB_W      176            // cols 0..175 hold w = -5..170 (zero padded)
#define SLAB_STRIDE 177            // odd -> LDS bank-conflict-free column reads

__global__ __launch_bounds__(32) void ssim_pass1_wblur_wmma(
    const float* __restrict__ img1, const float* __restrict__ img2,
    float* __restrict__ outA)
{
    __shared__ float s1[16 * SLAB_STRIDE];
    __shared__ float s2[16 * SLAB_STRIDE];

    const int blk   = blockIdx.x;          // 0..3199
    const int htile = blk % 10;
    const int d     = (blk / 10) % S;
    const int n     = blk / (10 * S);
    const int h0    = htile * 16;
    const int lane  = threadIdx.x;         // 0..31

    const size_t base_nd = (size_t)n * CUBE + (size_t)d * PLANE;

    // Fill LDS slabs (16*176 = 2816, exact multiple of 32 -> no divergence tail)
    for (int t = lane; t < 16 * SLAB_W; t += 32) {
        const int r = t / SLAB_W;
        const int p = t % SLAB_W;
        const int w = p - 5;               // conv pad = 5, zero padding
        float v1 = 0.0f, v2 = 0.0f;
        if (w >= 0 && w < S) {
            const size_t gi = base_nd + (size_t)(h0 + r) * ROWL + (size_t)w;
            v1 = img1[gi];
            v2 = img2[gi];
        }
        s1[r * SLAB_STRIDE + p] = v1;
        s2[r * SLAB_STRIDE + p] = v2;
    }
    __syncthreads();

    // Per-ISA 16x4 f32 A layout: lane L -> M = L&15; VGPR0 holds K = 2*(L>>4),
    // VGPR1 holds K = 2*(L>>4)+1. B (4x16) assumed symmetric: N = L&15,
    // VGPR j holds K row j + 2*(L>>4).
    const int m    = lane & 15;
    const int half = lane >> 4;

    // Banded Gaussian B fragments: KM[k][nn] = (0 <= k-nn <= 10) ? GK[k-nn] : 0
    v2f Bq[7];
#pragma unroll
    for (int q = 0; q < 7; ++q) {
        const int k0 = 4 * q + 2 * half;
        const int dx = k0 - m;
        const int dy = k0 + 1 - m;
        float bx = (dx >= 0 && dx <= 10) ? GK[dx] : 0.0f;
        float by = (dy >= 0 && dy <= 10) ? GK[dy] : 0.0f;
        Bq[q].x = bx;
        Bq[q].y = by;
    }

    for (int tc = 0; tc < 10; ++tc) {
        const int c0 = tc * 16;
        v8f acc0 = {}, acc1 = {}, acc2 = {}, acc3 = {}, acc4 = {};

#pragma unroll
        for (int q = 0; q < 7; ++q) {
            const int p = c0 + 4 * q + 2 * half;   // max 144+24+2+1 = 171 < 176
            const float a1x = s1[m * SLAB_STRIDE + p];
            const float a1y = s1[m * SLAB_STRIDE + p + 1];
            const float a2x = s2[m * SLAB_STRIDE + p];
            const float a2y = s2[m * SLAB_STRIDE + p + 1];

            v2f A0 = {a1x, a1y};                   // img1
            v2f A1 = {a2x, a2y};                   // img2
            v2f A2 = {a1x * a1x, a1y * a1y};       // img1^2
            v2f A3 = {a2x * a2x, a2y * a2y};       // img2^2
            v2f A4 = {a1x * a2x, a1y * a2y};       // img1*img2
            v2f b  = Bq[q];

            acc0 = __builtin_amdgcn_wmma_f32_16x16x4_f32(false, A0, false, b, (short)0, acc0, false, false);
            acc1 = __builtin_amdgcn_wmma_f32_16x16x4_f32(false, A1, false, b, (short)0, acc1, false, false);
            acc2 = __builtin_amdgcn_wmma_f32_16x16x4_f32(false, A2, false, b, (short)0, acc2, false, false);
            acc3 = __builtin_amdgcn_wmma_f32_16x16x4_f32(false, A3, false, b, (short)0, acc3, false, false);
            acc4 = __builtin_amdgcn_wmma_f32_16x16x4_f32(false, A4, false, b, (short)0, acc4, false, false);
        }

        // C/D layout: VGPR j -> M = j + 8*half, N = lane&15.
        const int col = c0 + m;
#pragma unroll
        for (int j = 0; j < 8; ++j) {
            const int row = j + 8 * half;
            const size_t gi = base_nd + (size_t)(h0 + row) * ROWL + (size_t)col;
            outA[gi]                   = acc0[j];
            outA[gi + (size_t)VOL]     = acc1[j];
            outA[gi + (size_t)2 * VOL] = acc2[j];
            outA[gi + (size_t)3 * VOL] = acc3[j];
            outA[gi + (size_t)4 * VOL] = acc4[j];
        }
    }
}

// ---------------------------------------------------------------------------
// Pass 2: H-axis blur. Thread = one w column (coalesced), 16-output register
// sliding window per (field, n, d, h-tile).
// ---------------------------------------------------------------------------
__global__ __launch_bounds__(160) void ssim_pass2_hblur(
    const float* __restrict__ A, float* __restrict__ B)
{
    const int w     = threadIdx.x;        // 0..159
    const int htile = blockIdx.x;         // 0..9
    const int d     = blockIdx.y;         // 0..159
    const int zf    = blockIdx.z;         // 0..9 = n*5 + f
    const int n     = zf / 5;
    const int f     = zf % 5;

    const size_t off = (size_t)f * VOL + (size_t)n * CUBE + (size_t)d * PLANE + (size_t)w;
    const float* src = A + off;
    float*       dst = B + off;
    const int h0 = htile * 16;

    float win[26];
#pragma unroll
    for (int j = 0; j < 26; ++j) {
        const int h = h0 - 5 + j;
        win[j] = (h >= 0 && h < S) ? src[(size_t)h * ROWL] : 0.0f;
    }
#pragma unroll
    for (int i = 0; i < 16; ++i) {
        float acc = 0.0f;
#pragma unroll
        for (int t = 0; t < 11; ++t) acc = fmaf(GK[t], win[i + t], acc);
        dst[(size_t)(h0 + i) * ROWL] = acc;
    }
}

// ---------------------------------------------------------------------------
// Pass 3: D-axis blur of all 5 fields + SSIM map + mean reduction (fp64 acc).
// ---------------------------------------------------------------------------
__device__ __forceinline__ void blur_d16(const float* __restrict__ p, int d0,
                                         float out[16])
{
    float win[26];
#pragma unroll
    for (int j = 0; j < 26; ++j) {
        const int d = d0 - 5 + j;
        win[j] = (d >= 0 && d < S) ? p[(size_t)d * PLANE] : 0.0f;
    }
#pragma unroll
    for (int i = 0; i < 16; ++i) {
        float acc = 0.0f;
#pragma unroll
        for (int t = 0; t < 11; ++t) acc = fmaf(GK[t], win[i + t], acc);
        out[i] = acc;
    }
}

__global__ __launch_bounds__(160) void ssim_pass3_dblur_reduce(
    const float* __restrict__ B, double* __restrict__ accum)
{
    const int w     = threadIdx.x;        // 0..159
    const int dtile = blockIdx.x;         // 0..9
    const int h     = blockIdx.y;         // 0..159
    const int n     = blockIdx.z;         // 0..1
    const int d0    = dtile * 16;

    const size_t off = (size_t)n * CUBE + (size_t)h * ROWL + (size_t)w;

    float mu1[16], mu2[16], b11[16], b22[16], b12[16];
    blur_d16(B + off,                   d0, mu1);
    blur_d16(B + off + (size_t)VOL,     d0, mu2);
    blur_d16(B + off + (size_t)2 * VOL, d0, b11);
    blur_d16(B + off + (size_t)3 * VOL, d0, b22);
    blur_d16(B + off + (size_t)4 * VOL, d0, b12);

    const float C1 = 1e-4f;   // (0.01 * 1.0)^2
    const float C2 = 9e-4f;   // (0.03 * 1.0)^2
    double lsum = 0.0;
#pragma unroll
    for (int i = 0; i < 16; ++i) {
        const float m1  = mu1[i], m2 = mu2[i];
        const float m1s = m1 * m1, m2s = m2 * m2, m12 = m1 * m2;
        const float sg1 = b11[i] - m1s;
        const float sg2 = b22[i] - m2s;
        const float sgx = b12[i] - m12;
        const float num = (2.0f * m12 + C1) * (2.0f * sgx + C2);
        const float den = (m1s + m2s + C1) * (sg1 + sg2 + C2);
        lsum += (double)(num / den);
    }

    __shared__ double red[160];
    red[w] = lsum;
    __syncthreads();
    if (w == 0) {
        double s = 0.0;
        for (int i = 0; i < 160; ++i) s += red[i];
        atomicAdd(accum, s);              // global_atomic_add_f64
    }
}

__global__ void ssim_finalize(const double* __restrict__ accum,
                              float* __restrict__ out)
{
    out[0] = (float)(accum[0] * (1.0 / (double)VOL));
}

// ---------------------------------------------------------------------------
extern "C" void kernel_launch(void* const* d_in, const int* in_sizes, int n_in,
                              void* d_out, int out_size, void* d_ws, size_t ws_size,
                              hipStream_t stream)
{
    const float* img1 = (const float*)d_in[0];
    const float* img2 = (const float*)d_in[1];
    float* out = (float*)d_out;

    float*  bufA  = (float*)d_ws;                        // 5 * VOL floats
    float*  bufB  = bufA + (size_t)5 * VOL;              // 5 * VOL floats
    double* accum = (double*)(bufB + (size_t)5 * VOL);   // 8-byte aligned

    hipMemsetAsync(accum, 0, sizeof(double), stream);

    ssim_pass1_wblur_wmma<<<dim3(NB * S * 10), dim3(32), 0, stream>>>(img1, img2, bufA);
    ssim_pass2_hblur<<<dim3(10, S, NB * 5), dim3(160), 0, stream>>>(bufA, bufB);
    ssim_pass3_dblur_reduce<<<dim3(10, S, NB), dim3(160), 0, stream>>>(bufB, accum);
    ssim_finalize<<<1, 1, 0, stream>>>(accum, out);
}